// ReliabilityDiagram_43946105373039
// MI455X (gfx1250) — compile-verified
//
#include <hip/hip_runtime.h>
#include <hip/hip_bf16.h>
#include <stdint.h>

// Reliability diagram / ECE for MI455X (gfx1250, wave32).
//
// Roofline: 33.5M x 8B = 268MB read-once -> ~11.5us @ 23.3 TB/s. Keep per
// element cost to ~14 VALU + 1 ds_add_u64 so HBM remains the bottleneck.
// All accumulation is integer (u64 packed fixed-point) -> order-independent
// -> bit-deterministic across graph replays.

#define NB_BINS 10
#define BLOCK_THREADS 320   // 10 waves of 32 (wave32)
#define REPLICAS 32         // one histogram replica per lane -> conflict-free LDS atomics
#define NWG_DEFAULT 2048    // partials footprint: 2048 * 32 * 4B = 256 KB of d_ws

typedef float v2f __attribute__((ext_vector_type(2)));
typedef float v4f __attribute__((ext_vector_type(4)));
typedef float v8f __attribute__((ext_vector_type(8)));
typedef int   v4i __attribute__((ext_vector_type(4)));

// u64 packing per (bin, lane-replica) slot:
//   bits [63:28] conf * 2^22 fixed point   (slot sum <= ~520 * 2^22 < 2^36)
//   bits [27:14] count                      (slot count <= ~520 < 2^14)
//   bits [13: 0] label sum                  (labels are 0/1)
#define CONF_SCALE 4194304.0f   /* 2^22 */
#define CONF_INV   (1.0f / 4194304.0f)

__device__ __forceinline__ float fast_sigmoid(float x) {
    float e = __expf(-x);                     // v_mul + v_exp_f32
    return __builtin_amdgcn_rcpf(1.0f + e);   // v_add + v_rcp_f32 (~1 ulp)
}

__device__ __forceinline__ unsigned long long pack_elem(float c, int label) {
    uint32_t cf = (uint32_t)__builtin_fmaf(c, CONF_SCALE, 0.5f); // round-to-nearest
    return ((unsigned long long)cf << 28) | 0x4000ull |
           (unsigned long long)((uint32_t)label & 1u);
}

__global__ __launch_bounds__(BLOCK_THREADS)
void ece_histogram_kernel(const float* __restrict__ logits,
                          const int*   __restrict__ labels,
                          float*       __restrict__ partials,  // [nWG][32] (30 used: bin*3 + {cnt,lab,conf})
                          int n)
{
    __shared__ unsigned long long acc[NB_BINS * REPLICAS]; // packed histogram
    __shared__ float fld[3][16][REPLICAS];                 // unpacked; rows 10..15 stay 0 for WMMA

    const int tid  = threadIdx.x;
    const int lane = tid & 31;

    // Zero LDS (320 u64 slots + 1536 floats).
    acc[tid] = 0ull;
    for (int i = tid; i < 3 * 16 * REPLICAS; i += BLOCK_THREADS)
        ((float*)fld)[i] = 0.0f;
    __syncthreads();

    const int gtid = blockIdx.x * BLOCK_THREADS + tid;
    const int T    = gridDim.x * BLOCK_THREADS;
    const int n4   = n >> 2;
    const v4f* __restrict__ lg4 = (const v4f*)logits;
    const v4i* __restrict__ lb4 = (const v4i*)labels;

    // Streaming loop: non-temporal b128 loads (268MB > 192MB L2; read-once).
    for (int i = gtid; i < n4; i += T) {
        v4f x = __builtin_nontemporal_load(&lg4[i]);
        v4i l = __builtin_nontemporal_load(&lb4[i]);
#pragma unroll
        for (int j = 0; j < 4; ++j) {
            float c = fast_sigmoid(x[j]);
            int b = (int)__builtin_ceilf(c * 10.0f) - 1;   // bin (b/10, (b+1)/10]
            b = min(max(b, 0), NB_BINS - 1);               // v_med3
            // One conflict-free LDS atomic per element: address bank = lane-distinct.
            atomicAdd(&acc[(b << 5) | lane], pack_elem(c, l[j]));
        }
    }
    // Scalar tail (N % 4) — no-op for the reference N.
    if (gtid == 0) {
        for (int k = n4 << 2; k < n; ++k) {
            float c = fast_sigmoid(logits[k]);
            int b = (int)__builtin_ceilf(c * 10.0f) - 1;
            b = min(max(b, 0), NB_BINS - 1);
            atomicAdd(&acc[(b << 5) | 0], pack_elem(c, labels[k]));
        }
    }
    __syncthreads();

    // Unpack each slot into float fields (all 320 threads, one slot each).
    {
        unsigned long long v = acc[tid];
        int b = tid >> 5, r = tid & 31;
        fld[0][b][r] = (float)(uint32_t)((v >> 14) & 0x3FFFull);        // count
        fld[1][b][r] = (float)(uint32_t)(v & 0x3FFFull);                // label sum
        fld[2][b][r] = (float)(uint32_t)(v >> 28) * CONF_INV;           // conf sum
    }
    __syncthreads();

    // Wave 0: reduce 32 replicas per bin with V_WMMA_F32_16X16X4_F32.
    // A (16x4, documented layout): lane m<16 holds A[m, 0..1] in {a.x,a.y},
    // lane m+16 holds A[m, 2..3]. B = all-ones (layout-independent since
    // uniform) -> D[m,n] = sum_k A[m,k], accumulated over 8 replica groups
    // via the C operand. 16 bin-rows; rows 10..15 are zeros.
    if ((tid >> 5) == 0) {
        const int bsel = lane & 15;
        const int h    = lane >> 4;      // 0: K={0,1}, 1: K={2,3}
        const v2f ones = {1.0f, 1.0f};
        v8f res0, res1, res2;
#pragma unroll
        for (int f = 0; f < 3; ++f) {
            v8f cacc = {};
#pragma unroll
            for (int g = 0; g < 8; ++g) {
                v2f a;
                a.x = fld[f][bsel][4 * g + 2 * h + 0];
                a.y = fld[f][bsel][4 * g + 2 * h + 1];
                cacc = __builtin_amdgcn_wmma_f32_16x16x4_f32(
                    /*neg_a=*/false, a, /*neg_b=*/false, ones,
                    /*c_mod=*/(short)0, cacc, /*reuse_a=*/false, /*reuse_b=*/false);
            }
            if (f == 0) res0 = cacc; else if (f == 1) res1 = cacc; else res2 = cacc;
        }
        // D layout: VGPR j, lanes 0-15 -> M=j ; lanes 16-31 -> M=j+8.
        float* base = partials + (size_t)blockIdx.x * 32;
        if (lane == 0) {
#pragma unroll
            for (int b = 0; b < 8; ++b) {
                base[b * 3 + 0] = res0[b];
                base[b * 3 + 1] = res1[b];
                base[b * 3 + 2] = res2[b];
            }
        } else if (lane == 16) {
#pragma unroll
            for (int b = 8; b < 10; ++b) {
                base[b * 3 + 0] = res0[b - 8];
                base[b * 3 + 1] = res1[b - 8];
                base[b * 3 + 2] = res2[b - 8];
            }
        }
    }
}

__global__ __launch_bounds__(256)
void ece_finalize_kernel(const float* __restrict__ partials, int nWG,
                         float* __restrict__ out)
{
    __shared__ float red[8][32];
    __shared__ float tot[32];
    const int tid   = threadIdx.x;
    const int pair  = tid & 31;   // 0..29 valid (bin*3 + field)
    const int slice = tid >> 5;

    float s = 0.0f;
    for (int wg = slice; wg < nWG; wg += 8)       // coalesced 128B rows
        s += partials[(size_t)wg * 32 + pair];
    red[slice][pair] = s;
    __syncthreads();

    if (tid < 32) {
        float t = 0.0f;
#pragma unroll
        for (int k = 0; k < 8; ++k) t += red[k][pair];
        tot[pair] = t;
    }
    __syncthreads();

    if (tid == 0) {
        float ece = 0.0f, mece = 0.0f;
        for (int b = 0; b < NB_BINS; ++b) {
            float cnt = tot[b * 3 + 0];
            float lab = tot[b * 3 + 1];
            float cf  = tot[b * 3 + 2];
            float pos = 0.0f, cavg = 0.0f, e = 0.0f;
            if (cnt > 0.0f) {
                float inv = 1.0f / cnt;
                pos  = lab * inv;
                cavg = cf * inv;
                e    = fabsf(pos - cavg);
            }
            out[b]      = pos;    // positives_per_bin
            out[10 + b] = cavg;   // confidence_per_bin
            ece += e;
            if (e > mece) mece = e;
        }
        out[20] = ece;
        out[21] = mece;
    }
}

extern "C" void kernel_launch(void* const* d_in, const int* in_sizes, int n_in,
                              void* d_out, int out_size, void* d_ws, size_t ws_size,
                              hipStream_t stream) {
    const float* logits = (const float*)d_in[0];
    const int*   labels = (const int*)d_in[1];
    const int n = in_sizes[0];

    float* partials = (float*)d_ws;
    int maxWG = (int)(ws_size / (32 * sizeof(float)));
    int nWG = NWG_DEFAULT;
    if (nWG > maxWG) nWG = maxWG;           // ws needs nWG*128B (256KB at default)
    if (nWG < 1) nWG = 1;

    ece_histogram_kernel<<<nWG, BLOCK_THREADS, 0, stream>>>(logits, labels, partials, n);
    ece_finalize_kernel<<<1, 256, 0, stream>>>(partials, nWG, (float*)d_out);
}